// TripletLoss_5652176961676
// MI455X (gfx1250) — compile-verified
//
#include <hip/hip_runtime.h>
#include <hip/hip_bf16.h>

#define B 512
#define D 256
#define MARGIN 1.0f

typedef __attribute__((ext_vector_type(2))) float v2f;
typedef __attribute__((ext_vector_type(8))) float v8f;

// ---------------------------------------------------------------------------
// Kernel 1: G = E * E^T via V_WMMA_F32_16X16X4_F32.
// One wave32 per 16x16 output tile; 64 WMMA steps over K=256.
//
// fp32 A-matrix 16x4 layout (ISA 7.12.2): lanes 0-15 hold K={0,1} in v[0],v[1];
// lanes 16-31 hold K={2,3}. M = lane&15. B (4x16) mirrors with N = lane&15.
// C/D: VGPR r, lane l -> M = r + 8*(l>>4), N = l&15.
// ---------------------------------------------------------------------------
__global__ __launch_bounds__(32) void gram_wmma(const float* __restrict__ E,
                                                float* __restrict__ G) {
    const int tile = blockIdx.x;          // 0..1023
    const int tm   = tile >> 5;           // row tile
    const int tn   = tile & 31;           // col tile
    const int lane = threadIdx.x;         // wave32
    const int half = lane >> 4;           // 0 or 1
    const int l15  = lane & 15;

    const float* __restrict__ arow = E + (tm * 16 + l15) * D + 2 * half;
    const float* __restrict__ brow = E + (tn * 16 + l15) * D + 2 * half;

    v8f c = {};
#pragma unroll 8
    for (int kk = 0; kk < D; kk += 4) {
        v2f a = *(const v2f*)(arow + kk);
        v2f b = *(const v2f*)(brow + kk);
        // 8 args: (neg_a, A, neg_b, B, c_mod, C, reuse_a, reuse_b)
        c = __builtin_amdgcn_wmma_f32_16x16x4_f32(
                false, a, false, b, (short)0, c, false, false);
    }

    const int row0 = tm * 16 + 8 * half;
    const int col  = tn * 16 + l15;
#pragma unroll
    for (int r = 0; r < 8; ++r)
        G[(row0 + r) * B + col] = c[r];
}

// ---------------------------------------------------------------------------
// Kernel 2: stash diag(G) into sq[] (before in-place transform), zero accums.
// ---------------------------------------------------------------------------
__global__ __launch_bounds__(512) void diag_init(const float* __restrict__ G,
                                                 float* __restrict__ sq,
                                                 float* __restrict__ acc) {
    const int t = threadIdx.x;
    sq[t] = G[t * B + t];
    if (t < 2) acc[t] = 0.0f;
}

// ---------------------------------------------------------------------------
// Kernel 3: pd[i,j] = (i==j) ? 0 : max(sq[i] - 2*G[i,j] + sq[j], 0)  (in place)
// ---------------------------------------------------------------------------
__global__ __launch_bounds__(256) void pd_fix(float* __restrict__ G,
                                              const float* __restrict__ sq) {
    const int idx = blockIdx.x * 256 + threadIdx.x;   // 0 .. B*B-1
    const int row = idx >> 9;
    const int col = idx & (B - 1);
    const float g = G[idx];
    float v = fmaxf(sq[row] - 2.0f * g + sq[col], 0.0f);
    G[idx] = (row == col) ? 0.0f : v;
}

// ---------------------------------------------------------------------------
// Kernel 4: semi-hard negative mining. One block (256 threads) per anchor j.
// row[k] reads inside the scan are uniform per iteration -> LDS broadcast.
// ---------------------------------------------------------------------------
__global__ __launch_bounds__(256) void semihard(const float* __restrict__ pd,
                                                const int* __restrict__ labels,
                                                float* __restrict__ acc) {
    __shared__ float row[B];
    __shared__ int   lab[B];
    __shared__ float red[256];

    const int j = blockIdx.x;
    const int t = threadIdx.x;

    row[t]       = pd[j * B + t];
    row[t + 256] = pd[j * B + t + 256];
    lab[t]       = labels[t];
    lab[t + 256] = labels[t + 256];
    __syncthreads();

    const int labj = lab[j];

    // ---- reductions over the anchor row: rowmax, rowmin, masked max -------
    const float v0 = row[t], v1 = row[t + 256];
    const bool  n0 = (lab[t] != labj), n1 = (lab[t + 256] != labj);

    // rowmax
    red[t] = fmaxf(v0, v1);
    __syncthreads();
    for (int s = 128; s > 0; s >>= 1) {
        if (t < s) red[t] = fmaxf(red[t], red[t + s]);
        __syncthreads();
    }
    const float rowmax = red[0];
    __syncthreads();

    // rowmin
    red[t] = fminf(v0, v1);
    __syncthreads();
    for (int s = 128; s > 0; s >>= 1) {
        if (t < s) red[t] = fminf(red[t], red[t + s]);
        __syncthreads();
    }
    const float rowmin = red[0];
    __syncthreads();

    // max over valid negatives (fallback base = -inf, clamp to rowmin later)
    float lm = -3.402823466e38f;
    if (n0) lm = fmaxf(lm, v0);
    if (n1) lm = fmaxf(lm, v1);
    red[t] = lm;
    __syncthreads();
    for (int s = 128; s > 0; s >>= 1) {
        if (t < s) red[t] = fmaxf(red[t], red[t + s]);
        __syncthreads();
    }
    // diag entry (same label) is always masked out -> rowmin participates
    const float neg_inside = fmaxf(red[0], rowmin);
    __syncthreads();

    // ---- per-positive scan: two candidates i = t and i = t+256 ------------
    const int   i0 = t, i1 = t + 256;
    const float t0 = row[i0], t1 = row[i1];
    const bool  pos0 = (lab[i0] == labj) && (i0 != j);
    const bool  pos1 = (lab[i1] == labj) && (i1 != j);

    float mn0 = rowmax, mn1 = rowmax;
    bool  any0 = false, any1 = false;
    for (int k = 0; k < B; ++k) {
        const float v   = row[k];          // broadcast read
        const bool  neg = (lab[k] != labj);
        if (neg && (v > t0)) { any0 = true; mn0 = fminf(mn0, v); }
        if (neg && (v > t1)) { any1 = true; mn1 = fminf(mn1, v); }
    }

    float lsum = 0.0f, lcnt = 0.0f;
    if (pos0) {
        const float semi = any0 ? mn0 : neg_inside;
        lsum += fmaxf(MARGIN + t0 - semi, 0.0f);
        lcnt += 1.0f;
    }
    if (pos1) {
        const float semi = any1 ? mn1 : neg_inside;
        lsum += fmaxf(MARGIN + t1 - semi, 0.0f);
        lcnt += 1.0f;
    }

    // ---- block reduce loss sum and positive count -------------------------
    red[t] = lsum;
    __syncthreads();
    for (int s = 128; s > 0; s >>= 1) {
        if (t < s) red[t] += red[t + s];
        __syncthreads();
    }
    const float bsum = red[0];
    __syncthreads();

    red[t] = lcnt;
    __syncthreads();
    for (int s = 128; s > 0; s >>= 1) {
        if (t < s) red[t] += red[t + s];
        __syncthreads();
    }
    if (t == 0) {
        atomicAdd(&acc[0], bsum);
        atomicAdd(&acc[1], red[0]);
    }
}

// ---------------------------------------------------------------------------
// Kernel 5: out = loss_sum / num_pos
// ---------------------------------------------------------------------------
__global__ void finalize(const float* __restrict__ acc, float* __restrict__ out) {
    out[0] = acc[0] / acc[1];
}

extern "C" void kernel_launch(void* const* d_in, const int* in_sizes, int n_in,
                              void* d_out, int out_size, void* d_ws, size_t ws_size,
                              hipStream_t stream) {
    const float* emb    = (const float*)d_in[0];   // [512, 256] fp32
    const int*   labels = (const int*)d_in[1];     // [512] int
    float*       out    = (float*)d_out;           // scalar fp32

    float* G   = (float*)d_ws;                     // 512*512 floats (Gram -> pd)
    float* sq  = G + B * B;                        // 512 floats
    float* acc = sq + B;                           // 2 floats (sum, count)

    gram_wmma<<<(B / 16) * (B / 16), 32, 0, stream>>>(emb, G);
    diag_init<<<1, B, 0, stream>>>(G, sq, acc);
    pd_fix<<<(B * B) / 256, 256, 0, stream>>>(G, sq);
    semihard<<<B, 256, 0, stream>>>(G, labels, acc);
    finalize<<<1, 1, 0, stream>>>(acc, out);
}